// ShowAttendTellCore_18794776888028
// MI455X (gfx1250) — compile-verified
//
#include <hip/hip_runtime.h>
#include <hip/hip_bf16.h>
#include <stdint.h>

// ---------------------------------------------------------------------------
// Sizes (compile-time, match reference)
// ---------------------------------------------------------------------------
#define BSZ      1024
#define DMODEL   512
#define NHEADS   8
#define NLEVELS  4
#define NPOINTS  4
#define DHEAD    64          // DMODEL / NHEADS
#define TTOTAL   1920        // 1024+512+256+128
#define RNN      512
#define ATTHID   512
#define LPP      16          // NLEVELS*NPOINTS

typedef __attribute__((ext_vector_type(2))) float v2f;
typedef __attribute__((ext_vector_type(8))) float v8f;

// ---------------------------------------------------------------------------
// Generic multi-segment fp32 WMMA GEMM, register-blocked 32x64 per wave:
//   C[m,n] = bias[n] + sum_s sum_k A_s[m,k] * B_s(k,n)
// A_s row-major (M x K_s). B_s(k,n) = TRANSB ? B_s[n*ldb_s + k]
//                                            : B_s[k*ldb_s + n]
// Per k-step: 2 A fragments + 4 B fragments feed 8 V_WMMA_F32_16X16X4_F32
// (2048 FLOP each) -> ~10.7 FLOP/byte of L1/L2 traffic per wave.
// Fragment layout (CDNA5 ISA 7.12.2):
//   A 16x4 : lane m=lane&15, VGPR{0,1} = K={kb,kb+1}, kb=2*(lane>=16)
//   B 4x16 : lane n=lane&15, VGPR{0,1} = K={kb,kb+1}
//   C 16x16: lane l, vgpr i -> (M = i + 8*(l>=16), N = l&15)
// ---------------------------------------------------------------------------
struct GemmArgs {
    const float* A[4];
    const float* B[4];
    int          K[4];
    int          ldb[4];
    int          nseg;
    const float* bias;   // may be null
    float*       C;
    int          M, N, ldc;
};

template <int TRANSB>
__global__ __launch_bounds__(256) void gemm_seg_wmma(GemmArgs ga) {
    const int lane = threadIdx.x & 31;
    const int wid  = threadIdx.x >> 5;
    const int ntn  = ga.N >> 6;                    // 64-col tiles
    const int w    = blockIdx.x * 8 + wid;         // flattened wave id
    const int tm   = w / ntn;
    const int tn   = w - tm * ntn;
    const int row0 = tm * 32;
    const int col0 = tn * 64;
    const int mn   = lane & 15;
    const int kb   = (lane >> 4) << 1;             // 0 or 2

    v8f c[2][4];
#pragma unroll
    for (int mi = 0; mi < 2; ++mi)
#pragma unroll
        for (int ni = 0; ni < 4; ++ni)
            c[mi][ni] = (v8f){0.f,0.f,0.f,0.f,0.f,0.f,0.f,0.f};

    for (int s = 0; s < ga.nseg; ++s) {
        const float* __restrict__ A  = ga.A[s];
        const float* __restrict__ Bm = ga.B[s];
        const int Ks  = ga.K[s];
        const int ldb = ga.ldb[s];
        const float* arow0 = A + (size_t)(row0 + mn) * Ks;
        const float* arow1 = arow0 + (size_t)16 * Ks;
        __builtin_prefetch(arow0, 0, 3);           // -> global_prefetch_b8
        __builtin_prefetch(arow1, 0, 3);
        if (TRANSB) {
            const float* br0 = Bm + (size_t)(col0 + mn) * ldb;
            const float* br1 = br0 + (size_t)16 * ldb;
            const float* br2 = br0 + (size_t)32 * ldb;
            const float* br3 = br0 + (size_t)48 * ldb;
            __builtin_prefetch(br0, 0, 3);
            for (int k = 0; k < Ks; k += 4) {
                v2f a0 = *(const v2f*)(arow0 + k + kb);
                v2f a1 = *(const v2f*)(arow1 + k + kb);
                v2f b0 = *(const v2f*)(br0 + k + kb);
                v2f b1 = *(const v2f*)(br1 + k + kb);
                v2f b2 = *(const v2f*)(br2 + k + kb);
                v2f b3 = *(const v2f*)(br3 + k + kb);
                c[0][0] = __builtin_amdgcn_wmma_f32_16x16x4_f32(false, a0, false, b0, (short)0, c[0][0], false, false);
                c[0][1] = __builtin_amdgcn_wmma_f32_16x16x4_f32(false, a0, false, b1, (short)0, c[0][1], false, false);
                c[0][2] = __builtin_amdgcn_wmma_f32_16x16x4_f32(false, a0, false, b2, (short)0, c[0][2], false, false);
                c[0][3] = __builtin_amdgcn_wmma_f32_16x16x4_f32(false, a0, false, b3, (short)0, c[0][3], false, false);
                c[1][0] = __builtin_amdgcn_wmma_f32_16x16x4_f32(false, a1, false, b0, (short)0, c[1][0], false, false);
                c[1][1] = __builtin_amdgcn_wmma_f32_16x16x4_f32(false, a1, false, b1, (short)0, c[1][1], false, false);
                c[1][2] = __builtin_amdgcn_wmma_f32_16x16x4_f32(false, a1, false, b2, (short)0, c[1][2], false, false);
                c[1][3] = __builtin_amdgcn_wmma_f32_16x16x4_f32(false, a1, false, b3, (short)0, c[1][3], false, false);
            }
        } else {
            const float* bc0 = Bm + col0 + mn;
            for (int k = 0; k < Ks; k += 4) {
                v2f a0 = *(const v2f*)(arow0 + k + kb);
                v2f a1 = *(const v2f*)(arow1 + k + kb);
                const float* bk0 = bc0 + (size_t)(k + kb) * ldb;
                const float* bk1 = bk0 + ldb;
                v2f b0, b1, b2, b3;
                b0[0] = bk0[0];  b0[1] = bk1[0];
                b1[0] = bk0[16]; b1[1] = bk1[16];
                b2[0] = bk0[32]; b2[1] = bk1[32];
                b3[0] = bk0[48]; b3[1] = bk1[48];
                c[0][0] = __builtin_amdgcn_wmma_f32_16x16x4_f32(false, a0, false, b0, (short)0, c[0][0], false, false);
                c[0][1] = __builtin_amdgcn_wmma_f32_16x16x4_f32(false, a0, false, b1, (short)0, c[0][1], false, false);
                c[0][2] = __builtin_amdgcn_wmma_f32_16x16x4_f32(false, a0, false, b2, (short)0, c[0][2], false, false);
                c[0][3] = __builtin_amdgcn_wmma_f32_16x16x4_f32(false, a0, false, b3, (short)0, c[0][3], false, false);
                c[1][0] = __builtin_amdgcn_wmma_f32_16x16x4_f32(false, a1, false, b0, (short)0, c[1][0], false, false);
                c[1][1] = __builtin_amdgcn_wmma_f32_16x16x4_f32(false, a1, false, b1, (short)0, c[1][1], false, false);
                c[1][2] = __builtin_amdgcn_wmma_f32_16x16x4_f32(false, a1, false, b2, (short)0, c[1][2], false, false);
                c[1][3] = __builtin_amdgcn_wmma_f32_16x16x4_f32(false, a1, false, b3, (short)0, c[1][3], false, false);
            }
        }
    }

    const int rb = (lane >> 4) << 3;               // 0 or 8
#pragma unroll
    for (int ni = 0; ni < 4; ++ni) {
        const int col = col0 + ni * 16 + mn;
        const float bias = ga.bias ? ga.bias[col] : 0.f;
#pragma unroll
        for (int mi = 0; mi < 2; ++mi) {
            float* Cp = ga.C + (size_t)(row0 + mi * 16 + rb) * ga.ldc + col;
#pragma unroll
            for (int i = 0; i < 8; ++i)
                Cp[(size_t)i * ga.ldc] = c[mi][ni][i] + bias;
        }
    }
}

static inline dim3 gemm_grid(int M, int N) {       // 8 waves/block, 32x64 tiles
    return dim3(((M / 32) * (N / 64)) / 8);
}

// ---------------------------------------------------------------------------
// Deformable sampling: softmax(aw) over 16 points, bilinear 1-D gather from v,
// fold attention weight -> clip_feats (B, H, 16, 64).
// One 64-thread block per (b,h); thread = d component.
// ---------------------------------------------------------------------------
__global__ __launch_bounds__(64) void sample_kernel(
        const float* __restrict__ v,        // (TTOTAL, DMODEL)
        const float* __restrict__ off_raw,  // (B, 128)
        const float* __restrict__ aw_raw,   // (B, 128)
        const float* __restrict__ refp,     // (B, NLEVELS)
        const int*   __restrict__ spatial,  // (NLEVELS)
        const int*   __restrict__ lstart,   // (NLEVELS)
        const unsigned char* __restrict__ padmask, // (TTOTAL) bool
        float* __restrict__ clip)           // (B*H*16, 64)
{
    const int bh = blockIdx.x;
    const int b  = bh >> 3;
    const int h  = bh & 7;
    const int d  = threadIdx.x;

    const float* awr  = aw_raw  + (size_t)b * 128 + h * LPP;
    const float* offr = off_raw + (size_t)b * 128 + h * LPP;

    float mx = -1e30f;
#pragma unroll
    for (int i = 0; i < LPP; ++i) mx = fmaxf(mx, awr[i]);
    float e[LPP], den = 0.f;
#pragma unroll
    for (int i = 0; i < LPP; ++i) { e[i] = __expf(awr[i] - mx); den += e[i]; }
    const float inv = 1.f / den;

#pragma unroll
    for (int lp = 0; lp < LPP; ++lp) {
        const int   l  = lp >> 2;
        const int   Ti = spatial[l];
        const float Tf = (float)Ti;
        const int   st = lstart[l];
        const float x  = refp[b * NLEVELS + l] * Tf + offr[lp] - 0.5f;
        const float x0f = floorf(x);
        const float w1  = x - x0f;
        const int   x0  = (int)x0f;
        const float valid0 = (x0 >= 0     && x0 < Ti)     ? 1.f : 0.f;
        const float valid1 = (x0 + 1 >= 0 && x0 + 1 < Ti) ? 1.f : 0.f;
        const int i0 = min(max(x0,     0), Ti - 1) + st;
        const int i1 = min(max(x0 + 1, 0), Ti - 1) + st;
        const float m0 = padmask[i0] ? 0.f : 1.f;
        const float m1 = padmask[i1] ? 0.f : 1.f;
        const float v0 = v[(size_t)i0 * DMODEL + h * DHEAD + d];
        const float v1 = v[(size_t)i1 * DMODEL + h * DHEAD + d];
        const float samp = v0 * ((1.f - w1) * valid0 * m0)
                         + v1 * (w1 * valid1 * m1);
        clip[((size_t)bh * LPP + lp) * DHEAD + d] = samp * (e[lp] * inv);
    }
}

// ---------------------------------------------------------------------------
// Fused additive-attention scorer:
//   dot[r] = b_alpha + sum_n w_alpha[n]*tanh( (clip@W_ctx)[r,n] + b_ctx[n]
//                                             + att_h[b(r),n] )
// 16 rows of one (b,h) per wave; A-fragments (16 k-steps) cached in VGPRs and
// reused across the 32 N-tiles; never materializes the 268MB 'att' tensor.
// ---------------------------------------------------------------------------
__global__ __launch_bounds__(256) void ctx_alpha_dot(
        const float* __restrict__ clip,     // (B*H*16, 64)
        const float* __restrict__ Wctx,     // (64, 512)
        const float* __restrict__ bctx,     // (512)
        const float* __restrict__ atth,     // (B, 512)
        const float* __restrict__ walpha,   // (512)
        const float* __restrict__ balpha,   // (1)
        float* __restrict__ dot)            // (B*H*16)
{
    const int lane = threadIdx.x & 31;
    const int wid  = threadIdx.x >> 5;
    const int tile = blockIdx.x * 8 + wid;   // 8192 tiles total
    const int r0   = tile * 16;              // rows of one (b,h)
    const int b    = r0 >> 7;                // 128 rows per batch element
    const int mn   = lane & 15;
    const int kb   = (lane >> 4) << 1;

    // Cache this wave's whole A tile (16x64) as WMMA fragments.
    v2f afrag[16];
    const float* arow = clip + (size_t)(r0 + mn) * DHEAD;
    __builtin_prefetch(arow, 0, 3);
#pragma unroll
    for (int s = 0; s < 16; ++s)
        afrag[s] = *(const v2f*)(arow + 4 * s + kb);

    float acc[8] = {0.f,0.f,0.f,0.f,0.f,0.f,0.f,0.f};

    for (int j = 0; j < 32; ++j) {           // N tiles of ATTHID=512
        const int n   = j * 16 + mn;
        const float bc = bctx[n];
        const float ah = atth[(size_t)b * ATTHID + n];
        const float wa = walpha[n];
        v8f c = {0.f,0.f,0.f,0.f,0.f,0.f,0.f,0.f};
#pragma unroll
        for (int s = 0; s < 16; ++s) {
            v2f bf;
            bf[0] = Wctx[(size_t)(4 * s + kb)     * ATTHID + n];
            bf[1] = Wctx[(size_t)(4 * s + kb + 1) * ATTHID + n];
            c = __builtin_amdgcn_wmma_f32_16x16x4_f32(
                    false, afrag[s], false, bf, (short)0, c, false, false);
        }
#pragma unroll
        for (int i = 0; i < 8; ++i)
            acc[i] += tanhf(c[i] + bc + ah) * wa;
    }

    // Reduce each accumulator across 16 lanes; halves stay independent so
    // lanes 0..15 end with rows 0..7 and lanes 16..31 with rows 8..15.
#pragma unroll
    for (int i = 0; i < 8; ++i) {
        float s = acc[i];
        s += __shfl_xor(s, 1, 32);
        s += __shfl_xor(s, 2, 32);
        s += __shfl_xor(s, 4, 32);
        s += __shfl_xor(s, 8, 32);
        acc[i] = s;
    }
    if (mn == 0) {                            // lanes 0 and 16
        const float ba = balpha[0];
        const int rb = r0 + ((lane >> 4) << 3);
#pragma unroll
        for (int i = 0; i < 8; ++i) dot[rb + i] = acc[i] + ba;
    }
}

// ---------------------------------------------------------------------------
// softmax over 16 points + weighted sum of clip_feats -> att_res (B, 512)
// ---------------------------------------------------------------------------
__global__ __launch_bounds__(64) void attres_kernel(
        const float* __restrict__ dot,   // (B*H*16)
        const float* __restrict__ clip,  // (B*H*16, 64)
        float* __restrict__ attres)      // (B, 512)
{
    const int bh = blockIdx.x;
    const int b  = bh >> 3;
    const int h  = bh & 7;
    const int d  = threadIdx.x;

    const float* dr = dot + (size_t)bh * LPP;
    float mx = -1e30f;
#pragma unroll
    for (int p = 0; p < LPP; ++p) mx = fmaxf(mx, dr[p]);
    float e[LPP], den = 0.f;
#pragma unroll
    for (int p = 0; p < LPP; ++p) { e[p] = __expf(dr[p] - mx); den += e[p]; }
    const float inv = 1.f / den;

    float s = 0.f;
#pragma unroll
    for (int p = 0; p < LPP; ++p)
        s += (e[p] * inv) * clip[((size_t)bh * LPP + p) * DHEAD + d];

    attres[(size_t)b * DMODEL + h * DHEAD + d] = s;
}

// ---------------------------------------------------------------------------
// LSTM pointwise: g = [i|f|g|o] gates (B,2048); c2 = sig(f)*c + sig(i)*tanh(g)
// ---------------------------------------------------------------------------
__global__ __launch_bounds__(256) void lstm_elem(
        const float* __restrict__ g,     // (B, 4*RNN)
        const float* __restrict__ cin,   // (B, RNN)
        float* __restrict__ hout,
        float* __restrict__ hout2,       // optional duplicate (may be null)
        float* __restrict__ cout)
{
    const int idx = blockIdx.x * 256 + threadIdx.x;   // < B*RNN
    const int b = idx >> 9;
    const int j = idx & 511;
    const float* gr = g + (size_t)b * (4 * RNN);
    const float gi = gr[j];
    const float gf = gr[RNN + j];
    const float gg = gr[2 * RNN + j];
    const float go = gr[3 * RNN + j];
    const float si = 1.f / (1.f + __expf(-gi));
    const float sf = 1.f / (1.f + __expf(-gf));
    const float so = 1.f / (1.f + __expf(-go));
    const float c2 = sf * cin[idx] + si * tanhf(gg);
    const float h2 = so * tanhf(c2);
    hout[idx] = h2;
    if (hout2) hout2[idx] = h2;
    cout[idx] = c2;
}

// ---------------------------------------------------------------------------
// Host-side orchestration
// ---------------------------------------------------------------------------
extern "C" void kernel_launch(void* const* d_in, const int* in_sizes, int n_in,
                              void* d_out, int out_size, void* d_ws, size_t ws_size,
                              hipStream_t stream) {
    const float* xt      = (const float*)d_in[0];
    const float* h0      = (const float*)d_in[1];   // (2,B,RNN)
    const float* c0      = (const float*)d_in[2];   // (2,B,RNN)
    const float* query   = (const float*)d_in[3];   // (1,B,DMODEL)
    const float* refp    = (const float*)d_in[4];   // (1,B,NLEVELS,1)
    const float* inflat  = (const float*)d_in[5];   // (1,TTOTAL,DMODEL)
    const int*   spatial = (const int*)d_in[6];
    const int*   lstart  = (const int*)d_in[7];
    const unsigned char* padmask = (const unsigned char*)d_in[8];
    const float* W_value = (const float*)d_in[9];
    const float* b_value = (const float*)d_in[10];
    const float* W_off   = (const float*)d_in[11];
    const float* b_off   = (const float*)d_in[12];
    const float* W_attw  = (const float*)d_in[13];
    const float* b_attw  = (const float*)d_in[14];
    const float* W_ctx   = (const float*)d_in[15];
    const float* b_ctx   = (const float*)d_in[16];
    const float* W_h2a   = (const float*)d_in[17];
    const float* b_h2a   = (const float*)d_in[18];
    const float* w_alpha = (const float*)d_in[19];
    const float* b_alpha = (const float*)d_in[20];
    const float* Wih0    = (const float*)d_in[21];  // (2048,1536)
    const float* Whh0    = (const float*)d_in[22];  // (2048,512)
    const float* Wih1    = (const float*)d_in[23];  // (2048,512)
    const float* Whh1    = (const float*)d_in[24];  // (2048,512)

    // workspace layout (floats)
    float* ws    = (float*)d_ws;
    float* v_ws  = ws;                 // 1920*512      =   983040
    float* offr  = ws + 983040;        // 1024*128      =   131072
    float* awr   = ws + 1114112;       // 1024*128
    float* clip  = ws + 1245184;       // 1024*8*16*64  =  8388608
    float* atth  = ws + 9633792;       // 1024*512
    float* dotb  = ws + 10158080;      // 1024*8*16
    float* ares  = ws + 10289152;      // 1024*512
    float* g0    = ws + 10813440;      // 1024*2048
    float* g1    = ws + 12910592;      // 1024*2048     (end 15007744 ~ 60MB)

    float* out      = (float*)d_out;   // [h2 | h1 | h2 | c1 | c2], each B*512
    float* h2_out   = out;
    float* h1_out   = out + 524288;
    float* h2_out_b = out + 1048576;
    float* c1_out   = out + 1572864;
    float* c2_out   = out + 2097152;

    const float* h_last = h0 + (size_t)BSZ * RNN;   // h0[-1] == h0[1]
    const float* h0l0   = h0;                       // h0[0]
    const float* c0l0   = c0;
    const float* c0l1   = c0 + (size_t)BSZ * RNN;

    // 1) value projection: v = input_flatten @ W_value + b_value   (1920x512)
    {
        GemmArgs a{};
        a.nseg = 1;
        a.A[0] = inflat; a.K[0] = DMODEL; a.B[0] = W_value; a.ldb[0] = DMODEL;
        a.bias = b_value; a.C = v_ws; a.M = TTOTAL; a.N = DMODEL; a.ldc = DMODEL;
        gemm_seg_wmma<0><<<gemm_grid(a.M, a.N), 256, 0, stream>>>(a);
    }
    // 2) off_raw = [h_last|query] @ W_off + b_off   (1024x128)
    {
        GemmArgs a{};
        a.nseg = 2;
        a.A[0] = h_last; a.K[0] = DMODEL; a.B[0] = W_off;             a.ldb[0] = 128;
        a.A[1] = query;  a.K[1] = DMODEL; a.B[1] = W_off + 512 * 128; a.ldb[1] = 128;
        a.bias = b_off; a.C = offr; a.M = BSZ; a.N = 128; a.ldc = 128;
        gemm_seg_wmma<0><<<gemm_grid(a.M, a.N), 256, 0, stream>>>(a);
    }
    // 3) aw_raw = [h_last|query] @ W_attw + b_attw  (1024x128)
    {
        GemmArgs a{};
        a.nseg = 2;
        a.A[0] = h_last; a.K[0] = DMODEL; a.B[0] = W_attw;             a.ldb[0] = 128;
        a.A[1] = query;  a.K[1] = DMODEL; a.B[1] = W_attw + 512 * 128; a.ldb[1] = 128;
        a.bias = b_attw; a.C = awr; a.M = BSZ; a.N = 128; a.ldc = 128;
        gemm_seg_wmma<0><<<gemm_grid(a.M, a.N), 256, 0, stream>>>(a);
    }
    // 4) att_h = h_last @ W_h2a + b_h2a   (1024x512)
    {
        GemmArgs a{};
        a.nseg = 1;
        a.A[0] = h_last; a.K[0] = RNN; a.B[0] = W_h2a; a.ldb[0] = ATTHID;
        a.bias = b_h2a; a.C = atth; a.M = BSZ; a.N = ATTHID; a.ldc = ATTHID;
        gemm_seg_wmma<0><<<gemm_grid(a.M, a.N), 256, 0, stream>>>(a);
    }
    // 5) deformable sampling -> clip_feats
    sample_kernel<<<BSZ * NHEADS, 64, 0, stream>>>(
        v_ws, offr, awr, refp, spatial, lstart, padmask, clip);
    // 6) fused tanh(clip@W_ctx + b + att_h) @ w_alpha -> dot
    ctx_alpha_dot<<<(BSZ * NHEADS * LPP / 16) / 8, 256, 0, stream>>>(
        clip, W_ctx, b_ctx, atth, w_alpha, b_alpha, dotb);
    // 7) softmax + weighted sum -> att_res
    attres_kernel<<<BSZ * NHEADS, 64, 0, stream>>>(dotb, clip, ares);
    // 8) LSTM0 gates: g0 = [xt|att_res|query]@Wih0^T + h0[0]@Whh0^T  (1024x2048)
    {
        GemmArgs a{};
        a.nseg = 4;
        a.A[0] = xt;    a.K[0] = 512; a.B[0] = Wih0;        a.ldb[0] = 1536;
        a.A[1] = ares;  a.K[1] = 512; a.B[1] = Wih0 + 512;  a.ldb[1] = 1536;
        a.A[2] = query; a.K[2] = 512; a.B[2] = Wih0 + 1024; a.ldb[2] = 1536;
        a.A[3] = h0l0;  a.K[3] = 512; a.B[3] = Whh0;        a.ldb[3] = 512;
        a.bias = nullptr; a.C = g0; a.M = BSZ; a.N = 4 * RNN; a.ldc = 4 * RNN;
        gemm_seg_wmma<1><<<gemm_grid(a.M, a.N), 256, 0, stream>>>(a);
    }
    // 9) LSTM0 pointwise -> h1, c1
    lstm_elem<<<(BSZ * RNN) / 256, 256, 0, stream>>>(g0, c0l0, h1_out, nullptr, c1_out);
    // 10) LSTM1 gates: g1 = h1@Wih1^T + h0[1]@Whh1^T  (1024x2048)
    {
        GemmArgs a{};
        a.nseg = 2;
        a.A[0] = h1_out; a.K[0] = 512; a.B[0] = Wih1; a.ldb[0] = 512;
        a.A[1] = h_last; a.K[1] = 512; a.B[1] = Whh1; a.ldb[1] = 512;
        a.bias = nullptr; a.C = g1; a.M = BSZ; a.N = 4 * RNN; a.ldc = 4 * RNN;
        gemm_seg_wmma<1><<<gemm_grid(a.M, a.N), 256, 0, stream>>>(a);
    }
    // 11) LSTM1 pointwise -> h2 (twice: tuple slot 0 and h_new[1]), c2
    lstm_elem<<<(BSZ * RNN) / 256, 256, 0, stream>>>(g1, c0l1, h2_out_b, h2_out, c2_out);
}